// TripletLoss_31662498906290
// MI455X (gfx1250) — compile-verified
//
#include <hip/hip_runtime.h>
#include <hip/hip_bf16.h>

typedef __attribute__((ext_vector_type(2))) float v2f;
typedef __attribute__((ext_vector_type(8))) float v8f;

#define B_SZ 384
#define D_SZ 256
#define MARGIN_F 0.3f

// ---------------------------------------------------------------------------
// Kernel 1: per-row squared norms (sq[i] = ||e_i||^2) and zero the accumulators.
// One block of 384 threads; float4 loads (row base is 1KB aligned).
// ---------------------------------------------------------------------------
__global__ __launch_bounds__(B_SZ) void sq_init_kernel(const float* __restrict__ E,
                                                       float* __restrict__ sq,
                                                       float* __restrict__ accum) {
  const int i = threadIdx.x;
  if (i < 2) accum[i] = 0.0f;
  if (i < B_SZ) {
    const float4* row = (const float4*)(E + (size_t)i * D_SZ);
    float s = 0.0f;
#pragma unroll 8
    for (int k = 0; k < D_SZ / 4; ++k) {
      float4 v = row[k];
      s = fmaf(v.x, v.x, fmaf(v.y, v.y, fmaf(v.z, v.z, fmaf(v.w, v.w, s))));
    }
    sq[i] = s;
  }
}

// ---------------------------------------------------------------------------
// Kernel 2: Gram matrix via V_WMMA_F32_16X16X4_F32, fused into squared
// distances:  dist[i][j] = max(sq[i] + sq[j] - 2*G[i][j], 0).
// One wave32 per 16x16 output tile.
//
// f32 16x16x4 operand layout (wave32):
//   A (16x4): lane L holds A[M = L%16][K = 2*(L/16) + {0,1}] in v[0],v[1]
//   B (4x16): lane L holds B[K = 2*(L/16) + {0,1}][N = L%16] in v[0],v[1]
// Since B = (E block J)^T for the Gram product, both operands are the same
// float2 gather from E, parameterized by the block index.
//   C/D (16x16 f32, 8 VGPRs): vgpr r: lanes 0-15 -> M=r, lanes 16-31 -> M=r+8,
//   N = lane % 16.
// ---------------------------------------------------------------------------
__global__ __launch_bounds__(32) void gram_dist_kernel(const float* __restrict__ E,
                                                       const float* __restrict__ sq,
                                                       float* __restrict__ dist) {
  const int I    = blockIdx.x;          // row-tile
  const int J    = blockIdx.y;          // col-tile
  const int lane = threadIdx.x & 31;
  const int half = lane >> 4;           // 0: lanes 0-15, 1: lanes 16-31
  const int l    = lane & 15;
  const int kb   = half * 2;            // K sub-offset for this half-wave

  const float* __restrict__ arow = E + (size_t)(I * 16 + l) * D_SZ;
  const float* __restrict__ brow = E + (size_t)(J * 16 + l) * D_SZ;

  v8f acc = {};                         // C = 0
#pragma unroll 8
  for (int k = 0; k < D_SZ; k += 4) {
    v2f a = *(const v2f*)(arow + k + kb);   // 8B-aligned
    v2f b = *(const v2f*)(brow + k + kb);
    // (neg_a, A, neg_b, B, c_mod, C, reuse_a, reuse_b)
    acc = __builtin_amdgcn_wmma_f32_16x16x4_f32(false, a, false, b,
                                                (short)0, acc, false, false);
  }

  const float sqj = sq[J * 16 + l];
#pragma unroll
  for (int r = 0; r < 8; ++r) {
    const int m   = r + 8 * half;
    const int row = I * 16 + m;
    const float d = fmaf(-2.0f, acc[r], sq[row] + sqj);
    dist[(size_t)row * B_SZ + J * 16 + l] = fmaxf(d, 0.0f);
  }
}

// ---------------------------------------------------------------------------
// Kernel 3: semi-hard triplet mining, one block (256 thr) per anchor `a`.
// dist row + labels staged in LDS; negmin via shared atomicMin on float bits
// (valid ordering since dist >= 0, INF = 0x7f800000).
// ---------------------------------------------------------------------------
__global__ __launch_bounds__(256) void loss_kernel(const float* __restrict__ dist,
                                                   const int* __restrict__ labels,
                                                   float* __restrict__ accum) {
  __shared__ float    drow[B_SZ];
  __shared__ int      lab[B_SZ];
  __shared__ unsigned negmin_bits;
  __shared__ int      hasneg;
  __shared__ float    bsum, bcnt;

  const int a   = blockIdx.x;
  const int tid = threadIdx.x;
  const int nth = blockDim.x;

  if (tid == 0) {
    negmin_bits = 0x7f800000u;  // +inf
    hasneg = 0;
    bsum = 0.0f;
    bcnt = 0.0f;
  }
  for (int n = tid; n < B_SZ; n += nth) {
    drow[n] = dist[(size_t)a * B_SZ + n];
    lab[n]  = labels[n];
  }
  __syncthreads();

  const int la = lab[a];

  // per-anchor min over negatives
  float lmin = __builtin_inff();
  int   lneg = 0;
  for (int n = tid; n < B_SZ; n += nth) {
    if (lab[n] != la) { lneg = 1; lmin = fminf(lmin, drow[n]); }
  }
  if (lneg) {
    atomicMin(&negmin_bits, __float_as_uint(lmin));
    atomicOr(&hasneg, 1);
  }
  __syncthreads();

  const float negmin = __uint_as_float(negmin_bits);
  const int   hn     = hasneg;

  float lsum = 0.0f, lcnt = 0.0f;
  if (hn) {
    for (int p = tid; p < B_SZ; p += nth) {
      if (p == a || lab[p] != la) continue;     // pos_mask = eq & ~eye
      const float dap = drow[p];
      const float hi  = dap + MARGIN_F;
      float smin = __builtin_inff();
      for (int n = 0; n < B_SZ; ++n) {
        const float d = drow[n];
        if ((lab[n] != la) && (d > dap) && (d < hi)) smin = fminf(smin, d);
      }
      const float dan  = (smin < __builtin_inff()) ? smin : negmin;
      const float loss = fmaxf(dap - dan + MARGIN_F, 0.0f);
      lsum += loss;
      lcnt += 1.0f;
    }
  }
  atomicAdd(&bsum, lsum);
  atomicAdd(&bcnt, lcnt);
  __syncthreads();
  if (tid == 0 && bcnt > 0.0f) {
    atomicAdd(&accum[0], bsum);
    atomicAdd(&accum[1], bcnt);
  }
}

// ---------------------------------------------------------------------------
// Kernel 4: final scalar
// ---------------------------------------------------------------------------
__global__ void finalize_kernel(const float* __restrict__ accum, float* __restrict__ out) {
  out[0] = accum[0] / fmaxf(accum[1], 1.0f);
}

extern "C" void kernel_launch(void* const* d_in, const int* in_sizes, int n_in,
                              void* d_out, int out_size, void* d_ws, size_t ws_size,
                              hipStream_t stream) {
  (void)in_sizes; (void)n_in; (void)out_size; (void)ws_size;
  const float* E      = (const float*)d_in[0];
  const int*   labels = (const int*)d_in[1];
  float*       out    = (float*)d_out;

  char*  ws    = (char*)d_ws;
  float* dist  = (float*)ws;                                        // 384*384 f32
  float* sq    = (float*)(ws + (size_t)B_SZ * B_SZ * sizeof(float)); // 384 f32
  float* accum = sq + B_SZ;                                          // 2 f32

  sq_init_kernel<<<1, B_SZ, 0, stream>>>(E, sq, accum);

  dim3 grid(B_SZ / 16, B_SZ / 16);
  gram_dist_kernel<<<grid, 32, 0, stream>>>(E, sq, dist);

  loss_kernel<<<B_SZ, 256, 0, stream>>>(dist, labels, accum);

  finalize_kernel<<<1, 1, 0, stream>>>(accum, out);
}